// GNNReachabilityNet_44109314130516
// MI455X (gfx1250) — compile-verified
//
#include <hip/hip_runtime.h>
#include <hip/hip_bf16.h>
#include <stdint.h>

typedef __attribute__((ext_vector_type(2))) float v2f;
typedef __attribute__((ext_vector_type(8))) float v8f;

static __device__ __forceinline__ void atomic_add_f32(float* p, float v) {
    __hip_atomic_fetch_add(p, v, __ATOMIC_RELAXED, __HIP_MEMORY_SCOPE_AGENT);
}

// ---------------------------------------------------------------- degree
__global__ void k_deg(const int* __restrict__ ei, float* __restrict__ deg, int nE) {
    int t = blockIdx.x * blockDim.x + threadIdx.x;
    if (t < nE) atomic_add_f32(&deg[ei[nE + t]], 1.0f);
}

// ---------------------------------------------------------------- f = x @ Wv + bv   (K=11)
__global__ void k_feat(const float* __restrict__ x, const float* __restrict__ Wv,
                       const float* __restrict__ bv, float* __restrict__ f) {
    const int node = blockIdx.x;
    const int j = threadIdx.x;                 // 64 output features
    const float* xr = x + (size_t)node * 11;
    float acc = bv[j];
#pragma unroll
    for (int k = 0; k < 11; ++k) acc = fmaf(xr[k], Wv[k * 64 + j], acc);
    f[(size_t)node * 64 + j] = acc;
}

// ---------------------------------------------------------------- edge pass 1: g[dst] += f[src], gc[dst] += col[src]
__global__ void k_edge1(const int* __restrict__ ei, const float* __restrict__ f,
                        const float* __restrict__ x, float* __restrict__ g,
                        float* __restrict__ gc, int nE) {
    const int e = (int)((blockIdx.x * blockDim.x + threadIdx.x) >> 5);
    const int lane = threadIdx.x & 31;
    if (e >= nE) return;
    const int src = ei[e];
    const int dst = ei[nE + e];
    const float* fr = f + (size_t)src * 64;
    float* gr = g + (size_t)dst * 64;
    const v2f v = *(const v2f*)(fr + lane * 2);          // coalesced 256B row read
    atomic_add_f32(&gr[lane * 2 + 0], v.x);
    atomic_add_f32(&gr[lane * 2 + 1], v.y);
    if (lane < 2) atomic_add_f32(&gc[(size_t)dst * 2 + lane], x[(size_t)src * 11 + 9 + lane]);
}

// ---------------------------------------------------------------- edge pass 2: g2[dst] += xv[src]
__global__ void k_edge2(const int* __restrict__ ei, const float* __restrict__ xv,
                        float* __restrict__ g, int nE) {
    const int e = (int)((blockIdx.x * blockDim.x + threadIdx.x) >> 5);
    const int lane = threadIdx.x & 31;
    if (e >= nE) return;
    const int src = ei[e];
    const int dst = ei[nE + e];
    const v2f v = *(const v2f*)(xv + (size_t)src * 64 + lane * 2);
    float* gr = g + (size_t)dst * 64;
    atomic_add_f32(&gr[lane * 2 + 0], v.x);
    atomic_add_f32(&gr[lane * 2 + 1], v.y);
}

// ---------------------------------------------------------------- node update 1:
//  U = [s*f | inv*g | s*col | inv*gc] (16x132 LDS)  -> h1 = U@We1 + s*be1 (LDS only)
//  -> xv = h1@Wx + bx  (written to HBM/L2)
__global__ void __launch_bounds__(256) k_node1(
    const float* __restrict__ f, const float* __restrict__ g,
    const float* __restrict__ gc, const float* __restrict__ x,
    const float* __restrict__ deg,
    const float* __restrict__ We1, const float* __restrict__ be1,
    const float* __restrict__ Wx,  const float* __restrict__ bx,
    float* __restrict__ xv)
{
    __shared__ float U[16 * 132];
    __shared__ float H1[16 * 128];
    __shared__ float sS[16], sI[16];
    const int row0 = blockIdx.x * 16;
    const int t = threadIdx.x;

    if (t < 16) {
        float d = deg[row0 + t];
        sS[t] = (d > 0.0f) ? 1.0f : 0.0f;
        sI[t] = 1.0f / fmaxf(d, 1.0f);
    }
    __syncthreads();

    for (int idx = t; idx < 16 * 132; idx += 256) {
        const int m = idx / 132, k = idx % 132;
        const int node = row0 + m;
        float v;
        if      (k < 64)  v = sS[m] * f [(size_t)node * 64 + k];
        else if (k < 128) v = sI[m] * g [(size_t)node * 64 + (k - 64)];
        else if (k < 130) v = sS[m] * x [(size_t)node * 11 + 9 + (k - 128)];
        else              v = sI[m] * gc[(size_t)node * 2 + (k - 130)];
        U[idx] = v;
    }
    __syncthreads();

    const int wave = t >> 5, lane = t & 31;
    const int r = lane & 15, half = lane >> 4;

    {   // h1 tile: 8 waves x 16 cols = 128 cols, K = 132
        const int n0 = wave * 16;
        v8f acc = {0.f, 0.f, 0.f, 0.f, 0.f, 0.f, 0.f, 0.f};
        for (int k = 0; k < 132; k += 4) {
            v2f a, b;
            a.x = U[r * 132 + k + 2 * half];
            a.y = U[r * 132 + k + 2 * half + 1];
            b.x = We1[(k + 2 * half)     * 128 + n0 + r];
            b.y = We1[(k + 2 * half + 1) * 128 + n0 + r];
            acc = __builtin_amdgcn_wmma_f32_16x16x4_f32(false, a, false, b,
                                                        (short)0, acc, false, false);
        }
#pragma unroll
        for (int j = 0; j < 8; ++j) {
            const int m = j + 8 * half;
            H1[m * 128 + n0 + r] = acc[j] + sS[m] * be1[n0 + r];
        }
    }
    __syncthreads();

    if (wave < 4) {   // xv tile: 4 waves x 16 cols = 64 cols, K = 128
        const int n0 = wave * 16;
        v8f acc = {0.f, 0.f, 0.f, 0.f, 0.f, 0.f, 0.f, 0.f};
        for (int k = 0; k < 128; k += 4) {
            v2f a, b;
            a.x = H1[r * 128 + k + 2 * half];
            a.y = H1[r * 128 + k + 2 * half + 1];
            b.x = Wx[(k + 2 * half)     * 64 + n0 + r];
            b.y = Wx[(k + 2 * half + 1) * 64 + n0 + r];
            acc = __builtin_amdgcn_wmma_f32_16x16x4_f32(false, a, false, b,
                                                        (short)0, acc, false, false);
        }
#pragma unroll
        for (int j = 0; j < 8; ++j) {
            const int m = j + 8 * half;
            xv[(size_t)(row0 + m) * 64 + n0 + r] = acc[j] + bx[n0 + r];
        }
    }
}

// ---------------------------------------------------------------- node update 2:
//  V = [s*xv | inv*g2] (16x128)  -> h2 = V@We2 + s*be2 (LDS)
//  -> node_out = h2@Wo + bo -> keyed atomicMax per graph
__global__ void __launch_bounds__(128) k_node2(
    const float* __restrict__ xv, const float* __restrict__ g2,
    const float* __restrict__ deg, const int* __restrict__ batch,
    const float* __restrict__ We2, const float* __restrict__ be2,
    const float* __restrict__ Wo,  const float* __restrict__ bo,
    unsigned* __restrict__ keys)
{
    __shared__ float V[16 * 128];
    __shared__ float H2[16 * 64];
    __shared__ float sS[16], sI[16];
    const int row0 = blockIdx.x * 16;
    const int t = threadIdx.x;

    if (t < 16) {
        float d = deg[row0 + t];
        sS[t] = (d > 0.0f) ? 1.0f : 0.0f;
        sI[t] = 1.0f / fmaxf(d, 1.0f);
    }
    __syncthreads();

    for (int idx = t; idx < 16 * 128; idx += 128) {
        const int m = idx >> 7, k = idx & 127;
        const int node = row0 + m;
        V[idx] = (k < 64) ? sS[m] * xv[(size_t)node * 64 + k]
                          : sI[m] * g2[(size_t)node * 64 + (k - 64)];
    }
    __syncthreads();

    const int wave = t >> 5, lane = t & 31;
    const int r = lane & 15, half = lane >> 4;
    {   // h2 tile: 4 waves x 16 cols = 64 cols, K = 128
        const int n0 = wave * 16;
        v8f acc = {0.f, 0.f, 0.f, 0.f, 0.f, 0.f, 0.f, 0.f};
        for (int k = 0; k < 128; k += 4) {
            v2f a, b;
            a.x = V[r * 128 + k + 2 * half];
            a.y = V[r * 128 + k + 2 * half + 1];
            b.x = We2[(k + 2 * half)     * 64 + n0 + r];
            b.y = We2[(k + 2 * half + 1) * 64 + n0 + r];
            acc = __builtin_amdgcn_wmma_f32_16x16x4_f32(false, a, false, b,
                                                        (short)0, acc, false, false);
        }
#pragma unroll
        for (int j = 0; j < 8; ++j) {
            const int m = j + 8 * half;
            H2[m * 64 + n0 + r] = acc[j] + sS[m] * be2[n0 + r];
        }
    }
    __syncthreads();

    if (t < 16) {   // node_out + segment_max (order-preserving uint key)
        float acc = bo[0];
        for (int n = 0; n < 64; ++n) acc = fmaf(H2[t * 64 + n], Wo[n], acc);
        const unsigned b = __float_as_uint(acc);
        const unsigned key = (b & 0x80000000u) ? ~b : (b | 0x80000000u);
        atomicMax(&keys[batch[row0 + t]], key);
    }
}

// ---------------------------------------------------------------- decode segment-max keys
__global__ void k_final(const unsigned* __restrict__ keys, float* __restrict__ out) {
    const int i = threadIdx.x;
    const unsigned k = keys[i];
    const unsigned b = (k & 0x80000000u) ? (k & 0x7fffffffu) : ~k;
    out[i] = __uint_as_float(b);
}

// ----------------------------------------------------------------
extern "C" void kernel_launch(void* const* d_in, const int* in_sizes, int n_in,
                              void* d_out, int out_size, void* d_ws, size_t ws_size,
                              hipStream_t stream) {
    const float* x    = (const float*)d_in[0];
    const int*   ei   = (const int*)  d_in[1];
    const int*   bat  = (const int*)  d_in[2];
    const float* Wv   = (const float*)d_in[3];
    const float* bv   = (const float*)d_in[4];
    const float* We1  = (const float*)d_in[5];
    const float* be1  = (const float*)d_in[6];
    const float* Wx   = (const float*)d_in[7];
    const float* bx   = (const float*)d_in[8];
    const float* We2  = (const float*)d_in[9];
    const float* be2  = (const float*)d_in[10];
    const float* Wo   = (const float*)d_in[11];
    const float* bo   = (const float*)d_in[12];
    float* out = (float*)d_out;

    const int nN = in_sizes[0] / 11;   // 100000 (multiple of 16)
    const int nE = in_sizes[1] / 2;    // 1600000

    char* ws = (char*)d_ws;
    float*    deg  = (float*)   (ws + 0);                      //  0.4 MB
    unsigned* keys = (unsigned*)(ws + (size_t)(1u  << 20));    //  256 B
    float*    gc   = (float*)   (ws + (size_t)(2u  << 20));    //  0.8 MB
    float*    f    = (float*)   (ws + (size_t)(4u  << 20));    // 25.6 MB
    float*    g    = (float*)   (ws + (size_t)(30u << 20));    // 25.6 MB (reused as g2)
    float*    xv   = (float*)   (ws + (size_t)(56u << 20));    // 25.6 MB
    const size_t fbytes = (size_t)nN * 64 * sizeof(float);

    hipMemsetAsync(deg,  0, (size_t)nN * sizeof(float), stream);
    hipMemsetAsync(keys, 0, 64 * sizeof(unsigned), stream);
    hipMemsetAsync(gc,   0, (size_t)nN * 2 * sizeof(float), stream);
    hipMemsetAsync(g,    0, fbytes, stream);

    k_deg  <<<(nE + 255) / 256, 256, 0, stream>>>(ei, deg, nE);
    k_feat <<<nN, 64, 0, stream>>>(x, Wv, bv, f);
    k_edge1<<<(nE + 7) / 8, 256, 0, stream>>>(ei, f, x, g, gc, nE);
    k_node1<<<nN / 16, 256, 0, stream>>>(f, g, gc, x, deg, We1, be1, Wx, bx, xv);

    hipMemsetAsync(g, 0, fbytes, stream);               // reset for pass 2
    k_edge2<<<(nE + 7) / 8, 256, 0, stream>>>(ei, xv, g, nE);
    k_node2<<<nN / 16, 128, 0, stream>>>(xv, g, deg, bat, We2, be2, Wo, bo, keys);
    k_final<<<1, 64, 0, stream>>>(keys, out);
}